// CustomLoss_77111842832551
// MI455X (gfx1250) — compile-verified
//
#include <hip/hip_runtime.h>
#include <hip/hip_bf16.h>

typedef float v2f __attribute__((ext_vector_type(2)));
typedef float v8f __attribute__((ext_vector_type(8)));

#define BLOCK    256
#define NBLOCKS  4096
#define GRID_N   7
#define CELLS    49
#define ROW      539
#define LAMBDA_C 5.0f
#define LAMBDA_N 0.5f
#define INV7     0.14285714285714285f

// Deterministic wave32 block reduction through the matrix pipe:
// V_WMMA_F32_16X16X4_F32 with B == ones gives D[m][n] = sum_k A[m][k].
// Each lane contributes its partial in exactly one A slot (odd-K slots zeroed),
// so lane0's row-half sum + lane16's row-half sum == exact 32-lane total,
// independent of the precise A striping. Full f32 precision throughout.
__device__ __forceinline__ float wave_block_reduce(float acc, float* lds)
{
    v2f a; a.x = acc;  a.y = 0.0f;   // A: acc in K-even slot, 0 in K-odd
    v2f b; b.x = 1.0f; b.y = 1.0f;   // B: all ones
    v8f c = {};
    v8f d = __builtin_amdgcn_wmma_f32_16x16x4_f32(false, a, false, b, (short)0, c,
                                                  false, false);
    float s = (d[0] + d[1]) + (d[2] + d[3]) + (d[4] + d[5]) + (d[6] + d[7]);
    // lanes 0..15 all hold one half-total, lanes 16..31 the other half-total.
    const int tid  = threadIdx.x;
    const int wave = tid >> 5;
    const int lane = tid & 31;
    if ((lane & 15) == 0)
        lds[wave * 2 + (lane >> 4)] = s;
    __syncthreads();
    float total = 0.0f;
    if (tid == 0) {
#pragma unroll
        for (int i = 0; i < (BLOCK / 32) * 2; ++i) total += lds[i];
    }
    return total;  // valid on tid 0 only
}

__device__ __forceinline__ float cell_loss(const float* __restrict__ o,
                                           float4 t0, float4 t1, bool valid1,
                                           float xl, float yl)
{
    const float cw = 1.0f / 7.0f;
    float contrib  = 0.0f;
    float conf_sum = 0.0f;
    bool  one_any  = false;
#pragma unroll
    for (int j = 0; j < 2; ++j) {
        const float px = o[5 * j + 0], py = o[5 * j + 1];
        const float pw = o[5 * j + 2], ph = o[5 * j + 3];
        const float pc = o[5 * j + 4];
        const float tx = j ? t1.x : t0.x;
        const float ty = j ? t1.y : t0.y;
        const float tw = j ? t1.z : t0.z;
        const float th = j ? t1.w : t0.w;
        const bool  valid = j ? valid1 : true;

        // IoU (v_rcp_f32 instead of the precise-div expansion: ~1ulp, fine
        // for a 6.4M-term loss sum, and deterministic)
        const float ix = fabsf(fmaxf(px - 0.5f * pw, tx - 0.5f * tw) -
                               fminf(px + 0.5f * pw, tx + 0.5f * tw));
        const float iy = fabsf(fmaxf(py - 0.5f * ph, ty - 0.5f * th) -
                               fminf(py + 0.5f * ph, ty + 0.5f * th));
        const float inter = ix * iy;
        const float uni   = pw * ph + tw * th - inter;
        const float conf  = inter * __builtin_amdgcn_rcpf(uni);

        // coord term (raw v_sqrt_f32, no denorm/range fixup code)
        const float dx = tx - px, dy = ty - py;
        const float dw = __builtin_amdgcn_sqrtf(tw) - __builtin_amdgcn_sqrtf(pw);
        const float dh = __builtin_amdgcn_sqrtf(th) - __builtin_amdgcn_sqrtf(ph);
        const float coord = dx * dx + dy * dy + dw * dw + dh * dh;

        const float dcf   = conf - pc;
        const float obj   = LAMBDA_C * coord + dcf * dcf;
        const float noobj = LAMBDA_N * pc * pc;

        const bool in_x = (xl <= tx) && (tx <= xl + cw);
        const bool in_y = (yl <= ty) && (ty <= yl + cw);
        const bool one  = in_x && in_y && valid;

        contrib  += one ? obj : (valid ? noobj : 0.0f);
        conf_sum += valid ? conf : 0.0f;
        one_any   = one_any || one;
    }
    const float dcls = o[10] - conf_sum;
    contrib += one_any ? dcls * dcls : 0.0f;
    return contrib;
}

__global__ __launch_bounds__(BLOCK) void loss_partials_kernel(
    const float* __restrict__ outputs, const float* __restrict__ target,
    float* __restrict__ partials, int npairs)
{
    __shared__ float lds[(BLOCK / 32) * 2];
    float acc = 0.0f;
    for (int i = blockIdx.x * BLOCK + threadIdx.x; i < npairs;
         i += NBLOCKS * BLOCK) {
        const unsigned b = (unsigned)i / CELLS;          // magic-mul div
        const unsigned c = (unsigned)i - b * CELLS;
        const unsigned cy = c / GRID_N;                  // magic-mul div
        const unsigned cx = c - cy * GRID_N;
        const float*  o  = outputs + (size_t)b * ROW + (size_t)c * 11;
        const float4* tg = (const float4*)(target + (size_t)b * 8);
        const float4 t0 = tg[0];
        const float4 t1 = tg[1];
        const bool valid1 = (t1.x != -1.0f);
        const float xl = (float)cx * INV7;
        const float yl = (float)cy * INV7;
        acc += cell_loss(o, t0, t1, valid1, xl, yl);
    }
    // All 256 threads reach here; WMMA below runs with EXEC all-ones.
    const float total = wave_block_reduce(acc, lds);
    if (threadIdx.x == 0) partials[blockIdx.x] = total;
}

__global__ __launch_bounds__(BLOCK) void loss_final_kernel(
    const float* __restrict__ partials, float* __restrict__ out)
{
    __shared__ float lds[(BLOCK / 32) * 2];
    float acc = 0.0f;
    // Fixed strided order -> deterministic.
    for (int i = threadIdx.x; i < NBLOCKS; i += BLOCK) acc += partials[i];
    const float total = wave_block_reduce(acc, lds);
    if (threadIdx.x == 0) out[0] = total;
}

extern "C" void kernel_launch(void* const* d_in, const int* in_sizes, int n_in,
                              void* d_out, int out_size, void* d_ws, size_t ws_size,
                              hipStream_t stream)
{
    const float* outputs = (const float*)d_in[0];   // (B, 539) f32
    const float* target  = (const float*)d_in[1];   // (B, 2, 4) f32
    float* out      = (float*)d_out;                // scalar
    float* partials = (float*)d_ws;                 // NBLOCKS floats (16 KB)

    const int b      = in_sizes[0] / ROW;
    const int npairs = b * CELLS;

    loss_partials_kernel<<<NBLOCKS, BLOCK, 0, stream>>>(outputs, target,
                                                        partials, npairs);
    loss_final_kernel<<<1, BLOCK, 0, stream>>>(partials, out);
}